// MultiHeadAttention_43757126812300
// MI455X (gfx1250) — compile-verified
//
#include <hip/hip_runtime.h>
#include <hip/hip_bf16.h>
#include <math.h>

// MI455X / gfx1250 fused neighbor-attention kernel.
// Algebraic restructure:
//   scores[s,h,n] = x_k[n] . g[s,h],  g[s,h] = qp[s,h] @ Wk_h   (never build Kp)
//   out[s,h]      = Y[s,h] @ Wv_h^T,  Y[s,h] = sum_n attn*x_v[n] (never build Vp)
// => kernel is HBM-bound streaming k/v once (~0.7 GB @ 23.3 TB/s ~ 30us);
//    all GEMMs (qp, G, Z, fc) run on V_WMMA_F32_16X16X4_F32 (full fp32 precision).
// CDNA5 specifics: wave32 wavefronts, v_wmma f32 path, global_prefetch_b8 for the
// k tile (overlaps HBM fetch with WMMA phases), non-temporal loads for the
// single-use v stream (keeps weights/k resident in L2).

typedef float v2f __attribute__((ext_vector_type(2)));
typedef float v4f __attribute__((ext_vector_type(4)));
typedef float v8f __attribute__((ext_vector_type(8)));

#define BATCH   64
#define NSRC    512
#define NNGH    20
#define CIN     128
#define NHEAD   4
#define DHEAD   32
#define TILE_S  16            // nodes per workgroup
#define NTHR    256           // 8 wave32
#define LDSTR   132           // LDS row stride (breaks 64-bank conflicts)

__global__ __launch_bounds__(NTHR)
void mha_fused_kernel(const float* __restrict__ q,   const float* __restrict__ k,
                      const float* __restrict__ v,   const float* __restrict__ Wq,
                      const float* __restrict__ Wk,  const float* __restrict__ Wv,
                      const float* __restrict__ Wfc, const float* __restrict__ bfc,
                      const float* __restrict__ gamma, const float* __restrict__ beta,
                      float* __restrict__ out, float* __restrict__ attn_out)
{
    __shared__ __align__(16) float sQP[TILE_S * LDSTR];           // qp tile [16 x 128]
    __shared__ __align__(16) float sGY[NHEAD * TILE_S * LDSTR];   // G then Y [4*16 x 128]
    __shared__ __align__(16) float sZ [TILE_S * LDSTR];           // attn output pre-fc
    __shared__ __align__(16) float sFC[TILE_S * LDSTR];           // fc output pre-LN
    __shared__ float sS[TILE_S * NHEAD * NNGH];                   // scores / attn
    __shared__ float sMu[TILE_S], sRstd[TILE_S];

    const int tid  = threadIdx.x;
    const int wave = tid >> 5;          // 0..7
    const int lane = tid & 31;
    const int mrow = lane & 15;         // M index / N index within 16-tile
    const int kh   = lane >> 4;         // selects K pair {0,1} vs {2,3}
    const int node_base = blockIdx.x * TILE_S;

    const float* kb = k + (size_t)node_base * NNGH * CIN;
    const float* vb = v + (size_t)node_base * NNGH * CIN;

    // ---------- Phase 0: prefetch this block's k tile (320 rows = 1280 x 128B) ----------
    // global_prefetch_b8: no LOADcnt, pure latency overlap with the WMMA phases below.
    for (int l = tid; l < TILE_S * NNGH * CIN / 32; l += NTHR)
        __builtin_prefetch(kb + (size_t)l * 32, 0, 1);

    // ---------- Phase 1: QP[16x128] = Qtile @ Wq^T (wave w -> N-tile w) ----------
    {
        const float* arow = q  + (size_t)(node_base + mrow) * CIN;
        const float* brow = Wq + (size_t)(wave * 16 + mrow) * CIN;   // B[kk,n]=Wq[n,kk]
        v8f acc = {};
        for (int kk = 0; kk < CIN; kk += 4) {
            v2f a = *(const v2f*)(arow + kk + 2 * kh);
            v2f b = *(const v2f*)(brow + kk + 2 * kh);
            acc = __builtin_amdgcn_wmma_f32_16x16x4_f32(false, a, false, b,
                                                        (short)0, acc, false, false);
        }
        for (int r = 0; r < 8; ++r)
            sQP[(r + 8 * kh) * LDSTR + wave * 16 + mrow] = acc[r];
    }
    __syncthreads();

    // ---------- Phase 2: G_h[16x128] = QP_h[16x32] @ Wk_h[32x128] ----------
    {
        const int h = wave >> 1;                    // 2 waves per head
        for (int nt = 0; nt < 4; ++nt) {
            const int n0 = ((wave & 1) * 4 + nt) * 16;
            v8f acc = {};
            for (int kk = 0; kk < DHEAD; kk += 4) {
                v2f a = *(const v2f*)(sQP + mrow * LDSTR + h * DHEAD + kk + 2 * kh);
                v2f b;                              // B[kk,i] = Wk[h*32+kk, i]
                b.x = Wk[(size_t)(h * DHEAD + kk + 2 * kh + 0) * CIN + n0 + mrow];
                b.y = Wk[(size_t)(h * DHEAD + kk + 2 * kh + 1) * CIN + n0 + mrow];
                acc = __builtin_amdgcn_wmma_f32_16x16x4_f32(false, a, false, b,
                                                            (short)0, acc, false, false);
            }
            for (int r = 0; r < 8; ++r)
                sGY[(h * 16 + r + 8 * kh) * LDSTR + n0 + mrow] = acc[r];
        }
    }
    __syncthreads();

    // ---------- Phase 3: scores[s,h,n] = k_row(s*20+n) . G[s,h] ----------
    // idx -> (group, n): 20 adjacent lanes share (s,h) => G reads are LDS broadcasts.
    for (int idx = tid; idx < TILE_S * NHEAD * NNGH; idx += NTHR) {
        const int grp = idx / NNGH;                 // 0..63 = (s,h)
        const int n   = idx - grp * NNGH;
        const int s   = grp >> 2;
        const int h   = grp & 3;
        const float* kr = kb + (size_t)(s * NNGH + n) * CIN;
        const float* g  = sGY + (h * 16 + s) * LDSTR;
        float acc = 0.f;
        for (int i = 0; i < CIN; i += 4) {
            v4f kv = *(const v4f*)(kr + i);
            v4f gv = *(const v4f*)(g + i);
            acc += kv.x * gv.x + kv.y * gv.y + kv.z * gv.z + kv.w * gv.w;
        }
        sS[(s * NHEAD + h) * NNGH + n] = acc;
    }
    __syncthreads();

    // ---------- Phase 4: softmax over n (no temperature, per reference) ----------
    if (tid < TILE_S * NHEAD) {
        const int s = tid >> 2, h = tid & 3;
        float* sc = sS + (s * NHEAD + h) * NNGH;
        float m = sc[0];
        for (int n = 1; n < NNGH; ++n) m = fmaxf(m, sc[n]);
        float sum = 0.f;
        for (int n = 0; n < NNGH; ++n) { float e = expf(sc[n] - m); sc[n] = e; sum += e; }
        const float inv = 1.f / sum;
        float* ao = attn_out + ((size_t)(node_base + s) * NHEAD + h) * NNGH;
        for (int n = 0; n < NNGH; ++n) { float a = sc[n] * inv; sc[n] = a; ao[n] = a; }
    }
    __syncthreads();

    // ---------- Phase 5: Y[h][s][:] = sum_n attn[s,h,n] * v_row ----------
    // thread owns (s, 8-col chunk) for all 4 heads: each v element loaded exactly once,
    // coalesced, and non-temporal (single-use stream; don't pollute L2).
    {
        const int s = tid >> 4;
        const int c = tid & 15;
        float acc[NHEAD][8];
        for (int h = 0; h < NHEAD; ++h)
            for (int j = 0; j < 8; ++j) acc[h][j] = 0.f;
        const float* vr0 = vb + (size_t)(s * NNGH) * CIN + c * 8;
        for (int n = 0; n < NNGH; ++n) {
            float w[NHEAD];
            for (int h = 0; h < NHEAD; ++h) w[h] = sS[(s * NHEAD + h) * NNGH + n];
            v4f va = __builtin_nontemporal_load((const v4f*)(vr0 + (size_t)n * CIN));
            v4f vc = __builtin_nontemporal_load((const v4f*)(vr0 + (size_t)n * CIN + 4));
            for (int h = 0; h < NHEAD; ++h) {
                acc[h][0] += w[h] * va.x; acc[h][1] += w[h] * va.y;
                acc[h][2] += w[h] * va.z; acc[h][3] += w[h] * va.w;
                acc[h][4] += w[h] * vc.x; acc[h][5] += w[h] * vc.y;
                acc[h][6] += w[h] * vc.z; acc[h][7] += w[h] * vc.w;
            }
        }
        for (int h = 0; h < NHEAD; ++h) {
            float* yp = sGY + (h * 16 + s) * LDSTR + c * 8;   // overwrites G (dead)
            v4f o0 = {acc[h][0], acc[h][1], acc[h][2], acc[h][3]};
            v4f o1 = {acc[h][4], acc[h][5], acc[h][6], acc[h][7]};
            *(v4f*)yp = o0; *(v4f*)(yp + 4) = o1;
        }
    }
    __syncthreads();

    // ---------- Phase 6: Z_h[16x32] = Y_h[16x128] @ Wv_h^T ----------
    {
        const int h  = wave >> 1;
        const int n0 = (wave & 1) * 16;
        const float* brow = Wv + (size_t)(h * DHEAD + n0 + mrow) * CIN; // B[kk,n]=Wv[h*32+n,kk]
        v8f acc = {};
        for (int kk = 0; kk < CIN; kk += 4) {
            v2f a = *(const v2f*)(sGY + (h * 16 + mrow) * LDSTR + kk + 2 * kh);
            v2f b = *(const v2f*)(brow + kk + 2 * kh);
            acc = __builtin_amdgcn_wmma_f32_16x16x4_f32(false, a, false, b,
                                                        (short)0, acc, false, false);
        }
        for (int r = 0; r < 8; ++r)
            sZ[(r + 8 * kh) * LDSTR + h * DHEAD + n0 + mrow] = acc[r];
    }
    __syncthreads();

    // ---------- Phase 7: FC[16x128] = Z @ Wfc^T + b_fc (wave w -> N-tile w) ----------
    {
        const float* brow = Wfc + (size_t)(wave * 16 + mrow) * CIN;
        v8f acc = {};
        for (int kk = 0; kk < CIN; kk += 4) {
            v2f a = *(const v2f*)(sZ + mrow * LDSTR + kk + 2 * kh);
            v2f b = *(const v2f*)(brow + kk + 2 * kh);
            acc = __builtin_amdgcn_wmma_f32_16x16x4_f32(false, a, false, b,
                                                        (short)0, acc, false, false);
        }
        const float bias = bfc[wave * 16 + mrow];
        for (int r = 0; r < 8; ++r)
            sFC[(r + 8 * kh) * LDSTR + wave * 16 + mrow] = acc[r] + bias;
    }
    __syncthreads();

    // ---------- Phase 8: LayerNorm over 128 and store ----------
    if (tid < TILE_S) {
        const float* row = sFC + tid * LDSTR;
        float s1 = 0.f, s2 = 0.f;
        for (int i = 0; i < CIN; ++i) { float x = row[i]; s1 += x; s2 += x * x; }
        const float mu  = s1 * (1.f / CIN);
        const float var = s2 * (1.f / CIN) - mu * mu;
        sMu[tid]   = mu;
        sRstd[tid] = rsqrtf(var + 1e-5f);
    }
    __syncthreads();
    for (int idx = tid; idx < TILE_S * CIN; idx += NTHR) {
        const int s = idx >> 7, c = idx & 127;
        const float val = (sFC[s * LDSTR + c] - sMu[s]) * sRstd[s] * gamma[c] + beta[c];
        out[(size_t)(node_base + s) * CIN + c] = val;
    }
}

extern "C" void kernel_launch(void* const* d_in, const int* in_sizes, int n_in,
                              void* d_out, int out_size, void* d_ws, size_t ws_size,
                              hipStream_t stream)
{
    const float* q     = (const float*)d_in[0];
    const float* k     = (const float*)d_in[1];
    const float* v     = (const float*)d_in[2];
    const float* Wq    = (const float*)d_in[3];
    const float* Wk    = (const float*)d_in[4];
    const float* Wv    = (const float*)d_in[5];
    const float* Wfc   = (const float*)d_in[6];
    const float* bfc   = (const float*)d_in[7];
    const float* gamma = (const float*)d_in[8];
    const float* beta  = (const float*)d_in[9];

    float* out      = (float*)d_out;
    float* attn_out = out + (size_t)BATCH * NSRC * (NHEAD * DHEAD);  // 4,194,304

    const int n_nodes = BATCH * NSRC;            // 32768
    const int grid    = n_nodes / TILE_S;        // 2048
    mha_fused_kernel<<<grid, NTHR, 0, stream>>>(q, k, v, Wq, Wk, Wv, Wfc, bfc,
                                                gamma, beta, out, attn_out);
}